// TokenPruning_19662360281645
// MI455X (gfx1250) — compile-verified
//
#include <hip/hip_runtime.h>
#include <hip/hip_bf16.h>

typedef float v2f __attribute__((ext_vector_type(2)));
typedef float v8f __attribute__((ext_vector_type(8)));

#define B_DIM 8
#define S_DIM 4096
#define D_DIM 1024
#define NUM_KEEP 2867   // max(1, int(4096 * (1 - 0.3)))

// ---------------------------------------------------------------------------
// Kernel 1: per-token L2 norm via V_WMMA_F32_16X16X4_F32.
// One wave handles 16 tokens. A = squared elements (16x4 fp32 slice), B = ones
// (4x16), accumulated over K=1024 -> every column of the 16x16 accumulator
// holds the per-token sum of squares. Full fp32 precision in the matrix pipe.
// Block = 128 threads (4 waves) -> 64 tokens per block.
// ---------------------------------------------------------------------------
__global__ __launch_bounds__(128)
void prune_importance_wmma(const float* __restrict__ tokens,
                           float* __restrict__ importance) {
    const int lane    = threadIdx.x & 31;
    const int wave    = threadIdx.x >> 5;
    const int tokBase = blockIdx.x * 64 + wave * 16;   // 16 tokens for this wave

    // Lane -> (row, k-offset) per the 16x4 fp32 A layout:
    // lanes 0-15: M = lane,   v0=K+0, v1=K+1
    // lanes 16-31: M = lane-16, v0=K+2, v1=K+3
    const int mRow = lane & 15;
    const int kOff = (lane >> 4) << 1;

    const float* src = tokens + (size_t)(tokBase + mRow) * D_DIM + kOff;

    v2f ones;
    ones.x = 1.0f; ones.y = 1.0f;

    v8f c0 = {};
    v8f c1 = {};

    // K = 1024, 4 per WMMA, two accumulators to break the C->D RAW chain.
    #pragma unroll 4
    for (int k = 0; k < D_DIM; k += 8) {
        v2f a0 = *(const v2f*)(src + k);
        v2f a1 = *(const v2f*)(src + k + 4);
        a0.x *= a0.x; a0.y *= a0.y;
        a1.x *= a1.x; a1.y *= a1.y;
        c0 = __builtin_amdgcn_wmma_f32_16x16x4_f32(
                 false, a0, false, ones, (short)0, c0, false, false);
        c1 = __builtin_amdgcn_wmma_f32_16x16x4_f32(
                 false, a1, false, ones, (short)0, c1, false, false);
    }

    v8f c = c0 + c1;

    // C/D layout: lanes 0-15 vgpr v holds D[M=v][N=lane]; lanes 16-31 hold
    // D[M=v+8][N=lane-16]. All columns are equal; pick 16 unique (M) writers.
    if (lane < 8) {
        importance[tokBase + lane] = sqrtf(c[lane]);
    } else if (lane >= 16 && lane < 24) {
        importance[tokBase + 8 + (lane - 16)] = sqrtf(c[lane - 16]);
    }
}

// ---------------------------------------------------------------------------
// Kernel 2: per-batch-row exact top-k selection.
// Bitonic sort (descending) of 4096 keys = (norm_bits<<32) | (4095-idx) in
// LDS, exactly reproducing top_k tie-breaking (value desc, lower idx first).
// Then scatter keep flags + block prefix scan -> compacted output positions.
// One 1024-thread block per batch row.
// ---------------------------------------------------------------------------
__global__ __launch_bounds__(1024)
void prune_select(const float* __restrict__ importance,
                  float* __restrict__ mask_out,
                  int* __restrict__ keep,
                  int* __restrict__ pos) {
    __shared__ unsigned long long keys[S_DIM];   // 32 KB
    __shared__ int partial[1024];

    const int b   = blockIdx.x;
    const int tid = threadIdx.x;
    const float* imp = importance + (size_t)b * S_DIM;

    // Pack keys. Norms are >= 0 so their IEEE bits are monotone as u32.
    for (int i = tid; i < S_DIM; i += 1024) {
        unsigned int vb = __float_as_uint(imp[i]);
        keys[i] = ((unsigned long long)vb << 32) |
                  (unsigned long long)(unsigned int)(S_DIM - 1 - i);
    }
    __syncthreads();

    // Bitonic sort, descending.
    for (int k = 2; k <= S_DIM; k <<= 1) {
        for (int j = k >> 1; j > 0; j >>= 1) {
            for (int i = tid; i < S_DIM; i += 1024) {
                int ixj = i ^ j;
                if (ixj > i) {
                    unsigned long long a = keys[i];
                    unsigned long long c = keys[ixj];
                    bool up = ((i & k) == 0);
                    if (up ? (a < c) : (a > c)) {
                        keys[i]   = c;
                        keys[ixj] = a;
                    }
                }
            }
            __syncthreads();
        }
    }

    // Read sorted ranks into registers, then reuse the keys LDS as flag array.
    int myIdx[4], myKeep[4];
    #pragma unroll
    for (int r = 0; r < 4; ++r) {
        int i = tid + r * 1024;                        // rank
        unsigned long long kk = keys[i];
        myIdx[r]  = S_DIM - 1 - (int)(unsigned int)(kk & 0xFFFFFFFFull);
        myKeep[r] = (i < NUM_KEEP) ? 1 : 0;
    }
    __syncthreads();

    int* flags = (int*)keys;                           // alias (16 KB used)
    #pragma unroll
    for (int r = 0; r < 4; ++r) {
        flags[myIdx[r]] = myKeep[r];
        mask_out[(size_t)b * S_DIM + myIdx[r]] = myKeep[r] ? 1.0f : 0.0f;
    }
    __syncthreads();

    // Block-wide exclusive prefix sum over 4096 flags (4 per thread).
    const int base = tid * 4;
    int s0 = flags[base] + flags[base + 1] + flags[base + 2] + flags[base + 3];
    partial[tid] = s0;
    __syncthreads();
    for (int off = 1; off < 1024; off <<= 1) {
        int add = (tid >= off) ? partial[tid - off] : 0;
        __syncthreads();
        partial[tid] += add;
        __syncthreads();
    }
    int run = partial[tid] - s0;                       // exclusive offset
    #pragma unroll
    for (int jj = 0; jj < 4; ++jj) {
        int idx = base + jj;
        int f = flags[idx];
        keep[(size_t)b * S_DIM + idx] = f;
        pos[(size_t)b * S_DIM + idx] = run;            // valid when f==1
        run += f;
    }
}

// ---------------------------------------------------------------------------
// Kernel 3: compaction gather. One 256-thread block per token; kept tokens
// copy their 1024 floats as float4 (b128) to the compacted slot.
// ---------------------------------------------------------------------------
__global__ __launch_bounds__(256)
void prune_gather(const float* __restrict__ tokens,
                  const int* __restrict__ keep,
                  const int* __restrict__ pos,
                  float* __restrict__ out) {
    const int tok = blockIdx.x;                        // b*S + s
    if (!keep[tok]) return;
    const int b = tok >> 12;                           // /4096
    const int p = pos[tok];

    const float4* src = (const float4*)(tokens + (size_t)tok * D_DIM);
    float4* dst = (float4*)(out + ((size_t)b * NUM_KEEP + p) * D_DIM);
    dst[threadIdx.x] = src[threadIdx.x];               // 256 * 16B = 4 KB
}

extern "C" void kernel_launch(void* const* d_in, const int* in_sizes, int n_in,
                              void* d_out, int out_size, void* d_ws, size_t ws_size,
                              hipStream_t stream) {
    (void)in_sizes; (void)n_in; (void)out_size; (void)ws_size;

    const float* tokens = (const float*)d_in[0];

    float* out_tokens = (float*)d_out;                                   // (8,2867,1024)
    float* out_mask   = (float*)d_out + (size_t)B_DIM * NUM_KEEP * D_DIM; // (8,4096)

    char* ws = (char*)d_ws;
    float* importance = (float*)ws;                                   // 32768 f32
    int*   keep       = (int*)(ws + (size_t)B_DIM * S_DIM * 4);       // 32768 i32
    int*   pos        = (int*)(ws + (size_t)B_DIM * S_DIM * 8);       // 32768 i32

    // 1) norms: 32768 tokens / 64 per block = 512 blocks
    prune_importance_wmma<<<512, 128, 0, stream>>>(tokens, importance);

    // 2) per-row top-k selection
    prune_select<<<B_DIM, 1024, 0, stream>>>(importance, out_mask, keep, pos);

    // 3) compaction gather
    prune_gather<<<B_DIM * S_DIM, 256, 0, stream>>>(tokens, keep, pos, out_tokens);
}